// VectorQuantizer_2954937500042
// MI455X (gfx1250) — compile-verified
//
#include <hip/hip_runtime.h>

// CDNA5 / gfx1250 — wave32, WMMA f32 16x16x4, async LDS staging
typedef __attribute__((ext_vector_type(2))) float v2f;
typedef __attribute__((ext_vector_type(8))) float v8f;
typedef __attribute__((ext_vector_type(4))) int   v4i;

#define BN_ROWS   (64 * 4096)         // 262144 rows of ze
#define DDIM      64
#define KCODES    512
#define ROWS_PW   32                  // rows per wave (two 16-row WMMA tiles)
#define THREADS   256                 // 8 waves (wave32)
#define WAVES_PB  8
#define NTILES    (BN_ROWS / ROWS_PW)        // 8192
#define NBLOCKS   (NTILES / WAVES_PB)        // 1024
#define WPAD      66                  // LDS row stride: bank-conflict-free, 8B-aligned rows

#if __has_builtin(__builtin_amdgcn_global_load_async_to_lds_b128) && \
    __has_builtin(__builtin_amdgcn_s_wait_asynccnt)
#define USE_ASYNC_LDS 1
#else
#define USE_ASYNC_LDS 0
#endif

#define WMMA_F32X4(A, B, C) \
    __builtin_amdgcn_wmma_f32_16x16x4_f32(false, (A), false, (B), (short)0, (C), false, false)

__global__ __launch_bounds__(THREADS)
void vq_main(const float* __restrict__ ze, const float* __restrict__ W,
             float* __restrict__ out, float* __restrict__ blockLoss) {
    extern __shared__ float smem[];
    float* Wlds   = smem;                    // [512][66]
    float* wnorm  = smem + KCODES * WPAD;    // [512]
    float* lossSm = wnorm + KCODES;          // [8]

    const int tid = threadIdx.x;

    // ---- stage full codebook W into LDS (padded rows) ----
#if USE_ASYNC_LDS
    for (int i = tid; i < (KCODES * DDIM) / 4; i += THREADS) {
        const int r = (i * 4) / DDIM;
        const int c = (i * 4) % DDIM;
        __builtin_amdgcn_global_load_async_to_lds_b128(
            (v4i*)(W + i * 4),               // global source (16B chunk)
            (v4i*)&Wlds[r * WPAD + c],       // LDS destination
            0, 0);
    }
    __builtin_amdgcn_s_wait_asynccnt(0);
#else
    for (int i = tid; i < (KCODES * DDIM) / 4; i += THREADS) {
        const float4 v = ((const float4*)W)[i];
        const int r = (i * 4) / DDIM;
        const int c = (i * 4) % DDIM;
        float* dst = &Wlds[r * WPAD + c];
        dst[0] = v.x; dst[1] = v.y; dst[2] = v.z; dst[3] = v.w;
    }
#endif
    __syncthreads();

    // ---- per-codeword squared norms ----
    for (int r = tid; r < KCODES; r += THREADS) {
        float s = 0.f;
        const float* row = &Wlds[r * WPAD];
        #pragma unroll
        for (int c = 0; c < DDIM; ++c) s += row[c] * row[c];
        wnorm[r] = s;
    }
    __syncthreads();

    const int lane  = tid & 31;
    const int wave  = tid >> 5;
    const int nl    = lane & 15;     // row (A) / col (B,C) within 16
    const int khalf = lane >> 4;     // K half for A/B fragments
    const int rowBase = (blockIdx.x * WAVES_PB + wave) * ROWS_PW;

    // ---- A fragments: two 16-row tiles sharing B; fuse ||ze||^2 ----
    v2f a0[16], a1[16];
    float zn0 = 0.f, zn1 = 0.f;
    const float* zr0 = ze + (size_t)(rowBase + nl) * DDIM + khalf * 2;
    const float* zr1 = zr0 + 16 * DDIM;
    #pragma unroll
    for (int kc = 0; kc < 16; ++kc) {
        const float2 t0 = *(const float2*)(zr0 + kc * 4);
        const float2 t1 = *(const float2*)(zr1 + kc * 4);
        a0[kc].x = t0.x; a0[kc].y = t0.y; zn0 += t0.x * t0.x + t0.y * t0.y;
        a1[kc].x = t1.x; a1[kc].y = t1.y; zn1 += t1.x * t1.x + t1.y * t1.y;
    }
    zn0 += __shfl_xor(zn0, 16, 32);
    zn1 += __shfl_xor(zn1, 16, 32);

    float dmin0[8], dmin1[8];
    int   didx0[8], didx1[8];
    #pragma unroll
    for (int j = 0; j < 8; ++j) {
        dmin0[j] = 3.4e38f; didx0[j] = 0;
        dmin1[j] = 3.4e38f; didx1[j] = 0;
    }

    // ---- sweep all 512 codewords; two independent WMMA chains per tile ----
    for (int n0 = 0; n0 < KCODES / 16; ++n0) {
        const int n = n0 * 16 + nl;
        const float* brow = &Wlds[(size_t)n * WPAD + khalf * 2];
        v8f acc0 = {}, acc1 = {};
        #pragma unroll
        for (int kc = 0; kc < 16; ++kc) {
            const float2 t = *(const float2*)(brow + kc * 4);
            v2f b; b.x = t.x; b.y = t.y;
            acc0 = WMMA_F32X4(a0[kc], b, acc0);
            acc1 = WMMA_F32X4(a1[kc], b, acc1);
        }
        const float wn = wnorm[n];
        #pragma unroll
        for (int j = 0; j < 8; ++j) {
            const float d20 = wn - 2.0f * acc0[j];
            const float d21 = wn - 2.0f * acc1[j];
            const bool b0 = d20 < dmin0[j];          // strict '<' keeps lowest n on ties
            const bool b1 = d21 < dmin1[j];
            dmin0[j] = b0 ? d20 : dmin0[j];  didx0[j] = b0 ? n : didx0[j];
            dmin1[j] = b1 ? d21 : dmin1[j];  didx1[j] = b1 ? n : didx1[j];
        }
    }

    // ---- argmin across the 16 lanes of each half (lowest-index tie-break) ----
    #pragma unroll
    for (int j = 0; j < 8; ++j) {
        #pragma unroll
        for (int m = 1; m <= 8; m <<= 1) {
            const float od0 = __shfl_xor(dmin0[j], m, 32);
            const int   oi0 = __shfl_xor(didx0[j], m, 32);
            const bool  t0  = (od0 < dmin0[j]) || (od0 == dmin0[j] && oi0 < didx0[j]);
            dmin0[j] = t0 ? od0 : dmin0[j];
            didx0[j] = t0 ? oi0 : didx0[j];
            const float od1 = __shfl_xor(dmin1[j], m, 32);
            const int   oi1 = __shfl_xor(didx1[j], m, 32);
            const bool  t1  = (od1 < dmin1[j]) || (od1 == dmin1[j] && oi1 < didx1[j]);
            dmin1[j] = t1 ? od1 : dmin1[j];
            didx1[j] = t1 ? oi1 : didx1[j];
        }
    }

    // ---- gather zq rows from LDS + accumulate loss contribution ----
    float lossAcc = 0.f;
    #pragma unroll
    for (int r = 0; r < 16; ++r) {
        const int   c   = __shfl(didx0[r & 7], (r >> 3) << 4, 32);
        const float dm  = __shfl(dmin0[r & 7], (r >> 3) << 4, 32);
        const float znr = __shfl(zn0, r, 32);
        lossAcc += znr + dm;                         // == ||ze_r - w_c||^2
        const float2 wv = *(const float2*)&Wlds[(size_t)c * WPAD + lane * 2];
        *(float2*)&out[(size_t)(rowBase + r) * DDIM + lane * 2] = wv;
    }
    #pragma unroll
    for (int r = 0; r < 16; ++r) {
        const int   c   = __shfl(didx1[r & 7], (r >> 3) << 4, 32);
        const float dm  = __shfl(dmin1[r & 7], (r >> 3) << 4, 32);
        const float znr = __shfl(zn1, r, 32);
        lossAcc += znr + dm;
        const float2 wv = *(const float2*)&Wlds[(size_t)c * WPAD + lane * 2];
        *(float2*)&out[(size_t)(rowBase + 16 + r) * DDIM + lane * 2] = wv;
    }

    if (lane == 0) lossSm[wave] = lossAcc;
    __syncthreads();
    if (tid == 0) {
        float s = 0.f;
        for (int w = 0; w < WAVES_PB; ++w) s += lossSm[w];
        blockLoss[blockIdx.x] = s;
    }
}

__global__ __launch_bounds__(256)
void vq_loss_reduce(const float* __restrict__ blockLoss, float* __restrict__ out) {
    __shared__ float sm[256];
    float s = 0.f;
    for (int i = threadIdx.x; i < NBLOCKS; i += 256) s += blockLoss[i];  // fixed order
    sm[threadIdx.x] = s;
    __syncthreads();
    for (int off = 128; off > 0; off >>= 1) {
        if (threadIdx.x < off) sm[threadIdx.x] += sm[threadIdx.x + off];
        __syncthreads();
    }
    if (threadIdx.x == 0)
        out[(size_t)BN_ROWS * DDIM] = 2.0f * sm[0] / (float)((size_t)BN_ROWS * DDIM);
}

extern "C" void kernel_launch(void* const* d_in, const int* in_sizes, int n_in,
                              void* d_out, int out_size, void* d_ws, size_t ws_size,
                              hipStream_t stream) {
    const float* ze = (const float*)d_in[0];   // [64,4096,64] fp32
    const float* W  = (const float*)d_in[1];   // [512,64] fp32
    float* out       = (float*)d_out;          // zq_st [B,N,D] ++ loss scalar
    float* blockLoss = (float*)d_ws;           // NBLOCKS floats scratch

    const size_t shmem = (size_t)(KCODES * WPAD + KCODES + WAVES_PB) * sizeof(float);
    vq_main<<<NBLOCKS, THREADS, shmem, stream>>>(ze, W, out, blockLoss);
    vq_loss_reduce<<<1, 256, 0, stream>>>(blockLoss, out);
}